// FlowMatchingLoss_55233279427250
// MI455X (gfx1250) — compile-verified
//
#include <hip/hip_runtime.h>
#include <hip/hip_bf16.h>

// ---------------------------------------------------------------------------
// RLOO / entropy / diversity fused reduction for MI455X (gfx1250, wave32).
//
// Memory-bound: ~148 MB streamed once -> ~6.4 us floor at 23.3 TB/s.
// One thread per row b: 5x b128 loads (pCS), 16+1 b32 loads (samples, Y).
// All reductions on-chip: shfl_xor (wave32) -> LDS -> V_WMMA_F32_16X16X4_F32
// (exact f32 FMA, RNE) for the cross-wave block sum. Deterministic 2-pass
// reduction (no float atomics), so graph replays are bitwise-stable.
// ---------------------------------------------------------------------------

#define EPS_F 1e-8f

typedef __attribute__((ext_vector_type(2))) float v2f;
typedef __attribute__((ext_vector_type(8))) float v8f;

// Accumulator indices: 0 = sum p*log(p+eps) (both rows), 1 = unique-pair count,
// 2 = correct count, 3 = sum log_prob * advantage.

__launch_bounds__(256)
__global__ void rloo_partial_kernel(const float* __restrict__ pCS,
                                    const int*   __restrict__ Y,
                                    const int*   __restrict__ S1,
                                    const int*   __restrict__ S2,
                                    float*       __restrict__ partial,
                                    int B)
{
    // stride-21 padding: 21 is odd -> t*21 mod 64 distinct across a wave,
    // bank-conflict-free indexed gathers.
    __shared__ float slp[256 * 21];
    __shared__ float wred[8][4];

    const int tid = threadIdx.x;
    const int b   = blockIdx.x * 256 + tid;   // B == gridDim*256 exactly

    // ---- stream the 20 probabilities for this row (5x global_load_b128) ----
    const float4* pv = reinterpret_cast<const float4*>(pCS + (size_t)b * 20u);
    float p[20];
#pragma unroll
    for (int i = 0; i < 5; ++i) {
        float4 q = pv[i];
        p[4*i+0] = q.x; p[4*i+1] = q.y; p[4*i+2] = q.z; p[4*i+3] = q.w;
    }

    // ---- entropy terms + log table for gathers ----
    float* lp = &slp[tid * 21];
    float ent = 0.f;
#pragma unroll
    for (int k = 0; k < 20; ++k) {
        float pk = p[k];
        ent  += pk * __logf(pk + EPS_F);   // entropy uses log(p + eps)
        lp[k] = __logf(pk);                // RLOO uses log(p)
    }
    // lp[0..9] = log p1, lp[10..19] = log p2

    // ---- samples (coalesced b32 loads, stride B between s indices) ----
    int a1[8], a2[8];
#pragma unroll
    for (int s = 0; s < 8; ++s) {
        a1[s] = S1[(size_t)s * B + b];
        a2[s] = S2[(size_t)s * B + b];
    }
    const int y = Y[b];

    // ---- duplicate detection via 100-bit bitmap + correctness ----
    unsigned long long m0 = 0ull, m1 = 0ull;
    float corr[8];
    float csum = 0.f;
#pragma unroll
    for (int s = 0; s < 8; ++s) {
        int pr = a1[s] * 10 + a2[s];                 // pair id in [0,100)
        unsigned long long bit = 1ull << (pr & 63);
        if (pr < 64) m0 |= bit; else m1 |= bit;
        float c = (a1[s] + a2[s] == y) ? 1.f : 0.f;
        corr[s] = c; csum += c;
    }
    float uniq = (float)(__popcll(m0) + __popcll(m1));

    // ---- RLOO: advantage-weighted log-prob (LDS gathers) ----
    const float invs = 1.f / (7.f + 1e-8f);          // 1/(S-1+eps)
    float lpadv = 0.f;
#pragma unroll
    for (int s = 0; s < 8; ++s) {
        float adv = corr[s] - (csum - corr[s]) * invs;
        float l   = lp[a1[s]] + lp[10 + a2[s]];
        lpadv += l * adv;
    }

    // ---- wave32 butterfly reduction (ds_swizzle / permlane paths) ----
    float r0 = ent, r1 = uniq, r2 = csum, r3 = lpadv;
#pragma unroll
    for (int off = 16; off > 0; off >>= 1) {
        r0 += __shfl_xor(r0, off, 32);
        r1 += __shfl_xor(r1, off, 32);
        r2 += __shfl_xor(r2, off, 32);
        r3 += __shfl_xor(r3, off, 32);
    }
    const int wave = tid >> 5;
    const int lane = tid & 31;
    if (lane == 0) {
        wred[wave][0] = r0; wred[wave][1] = r1;
        wred[wave][2] = r2; wred[wave][3] = r3;
    }
    __syncthreads();

    // ---- cross-wave sum via V_WMMA_F32_16X16X4_F32 (wave 0, EXEC all-1s) ----
    // A (16x4 f32): row a = accumulator a, col k = wave k's partial.
    // Lanes 0-15 carry K=0,1 (VGPR comps x,y); lanes 16-31 carry K=2,3.
    // B = all-ones 4x16  =>  D[a][j] = sum_k A[a][k].  Two chained WMMAs
    // cover waves 0..3 then 4..7.  D[a][0] lives in lane 0, component a.
    if (tid < 32) {
        const int a  = tid & 15;
        const int w0 = (tid >> 4) * 2;      // lanes 0-15 -> waves {0,1}; 16-31 -> {2,3}
        const bool ok = (a < 4);
        v2f A1, A2, Bv;
        A1[0] = ok ? wred[w0 + 0][a] : 0.f;
        A1[1] = ok ? wred[w0 + 1][a] : 0.f;
        A2[0] = ok ? wred[w0 + 4][a] : 0.f;
        A2[1] = ok ? wred[w0 + 5][a] : 0.f;
        Bv[0] = 1.f; Bv[1] = 1.f;
        v8f c = {};
        c = __builtin_amdgcn_wmma_f32_16x16x4_f32(false, A1, false, Bv,
                                                  (short)0, c, false, false);
        c = __builtin_amdgcn_wmma_f32_16x16x4_f32(false, A2, false, Bv,
                                                  (short)0, c, false, false);
        if (tid == 0) {
            float* outp = partial + (size_t)blockIdx.x * 4u;
            outp[0] = c[0]; outp[1] = c[1]; outp[2] = c[2]; outp[3] = c[3];
        }
    }
}

__launch_bounds__(256)
__global__ void rloo_finalize_kernel(const float* __restrict__ partial,
                                     const float* __restrict__ base_loss,
                                     float*       __restrict__ out,
                                     int nblocks, float Bf)
{
    __shared__ float wred[8][4];
    float r0 = 0.f, r1 = 0.f, r2 = 0.f, r3 = 0.f;
    for (int i = threadIdx.x; i < nblocks; i += 256) {
        float4 q = reinterpret_cast<const float4*>(partial)[i];
        r0 += q.x; r1 += q.y; r2 += q.z; r3 += q.w;
    }
#pragma unroll
    for (int off = 16; off > 0; off >>= 1) {
        r0 += __shfl_xor(r0, off, 32);
        r1 += __shfl_xor(r1, off, 32);
        r2 += __shfl_xor(r2, off, 32);
        r3 += __shfl_xor(r3, off, 32);
    }
    const int wave = threadIdx.x >> 5;
    const int lane = threadIdx.x & 31;
    if (lane == 0) {
        wred[wave][0] = r0; wred[wave][1] = r1;
        wred[wave][2] = r2; wred[wave][3] = r3;
    }
    __syncthreads();

    if (threadIdx.x == 0) {
        float E = 0.f, U = 0.f, C = 0.f, L = 0.f;
#pragma unroll
        for (int w = 0; w < 8; ++w) {
            E += wred[w][0]; U += wred[w][1];
            C += wred[w][2]; L += wred[w][3];
        }
        // ent1+ent2 (means) = -E/B ; entropy_loss = -(ent1+ent2)/2 = E/(2B)
        float entropy_loss   = E / (2.f * Bf);
        float ratio          = U / (8.f * Bf);            // diversity / max_diversity
        float diversity_loss = -__logf(ratio + EPS_F);
        float sample_acc     = C / (8.f * Bf);
        float rloo_loss      = -L / (8.f * Bf);
        float total = base_loss[0] + 0.01f * entropy_loss + 0.1f * diversity_loss;
        out[0] = total;
        out[1] = rloo_loss;
        out[2] = diversity_loss;
        out[3] = sample_acc;
    }
}

extern "C" void kernel_launch(void* const* d_in, const int* in_sizes, int n_in,
                              void* d_out, int out_size, void* d_ws, size_t ws_size,
                              hipStream_t stream) {
    const float* pCS       = (const float*)d_in[0];   // (B, 2, 10) f32
    const float* base_loss = (const float*)d_in[1];   // (1,)       f32
    const int*   Y         = (const int*)d_in[2];     // (B,)       i32
    const int*   S1        = (const int*)d_in[3];     // (8, B)     i32
    const int*   S2        = (const int*)d_in[4];     // (8, B)     i32

    const int B = in_sizes[2];                        // 1048576 (multiple of 256)
    const int nblocks = B / 256;

    float* partial = (float*)d_ws;                    // nblocks*4 floats, fully
                                                      // rewritten every launch

    rloo_partial_kernel<<<nblocks, 256, 0, stream>>>(pCS, Y, S1, S2, partial, B);
    rloo_finalize_kernel<<<1, 256, 0, stream>>>(partial, base_loss,
                                                (float*)d_out, nblocks, (float)B);
}